// GNNUS_BaseModel_16432544874724
// MI455X (gfx1250) — compile-verified
//
#include <hip/hip_runtime.h>
#include <math.h>

typedef float v2f __attribute__((ext_vector_type(2)));
typedef float v8f __attribute__((ext_vector_type(8)));

#define BG 64          // batch graphs
#define NN 512         // nodes per graph
#define NC 7           // classes
#define NH 20          // ARMA hidden

// ---------------------------------------------------------------------------
// K1: dinv[m][b][j] = rsqrt(max(sum_i A_m[b,i,j], eps)) (0 if deg<=0)
// ---------------------------------------------------------------------------
__global__ void k_dinv(const float* __restrict__ A0, const float* __restrict__ A1,
                       const float* __restrict__ A2, const float* __restrict__ A3,
                       float* __restrict__ dinv) {
    int idx = blockIdx.x * blockDim.x + threadIdx.x;        // [0, 4*BG*NN)
    int m   = idx / (BG * NN);
    int rem = idx % (BG * NN);
    int b   = rem / NN;
    int j   = rem % NN;
    const float* A = (m == 0) ? A0 : (m == 1) ? A1 : (m == 2) ? A2 : A3;
    const float* col = A + (size_t)b * NN * NN + j;
    float s = 0.f;
    #pragma unroll 4
    for (int i = 0; i < NN; ++i) s += col[(size_t)i * NN];
    dinv[idx] = (s > 0.f) ? rsqrtf(fmaxf(s, 1e-12f)) : 0.f;
}

// ---------------------------------------------------------------------------
// K2 (small f): scalar feature transform.
//   S[row,h] = dinv[row] * (x[row,:] @ Wi[:,h])   (zero-padded to strideOut)
//   R[row,h] = x[row,:] @ Wr[:,h]
// ---------------------------------------------------------------------------
__global__ void k_xform(const float* __restrict__ x, const float* __restrict__ Wi,
                        const float* __restrict__ Wr, const float* __restrict__ dinv,
                        float* __restrict__ S, float* __restrict__ R,
                        int f, int Hout, int strideOut) {
    int idx = blockIdx.x * blockDim.x + threadIdx.x;
    int row = idx / strideOut;            // b*NN + i
    int h   = idx % strideOut;
    if (row >= BG * NN) return;
    if (h >= Hout) { S[idx] = 0.f; R[idx] = 0.f; return; }
    const float* xr = x + (size_t)row * f;
    float ai = 0.f, ar = 0.f;
    #pragma unroll 4
    for (int k = 0; k < f; ++k) {
        float xv = xr[k];
        ai += xv * Wi[k * Hout + h];
        ar += xv * Wr[k * Hout + h];
    }
    S[idx] = dinv[row] * ai;
    R[idx] = ar;
}

// ---------------------------------------------------------------------------
// K2b: pad/concat weights for the WMMA transform:
//   Wcat[k][0..19]  = Wi[k][:],  Wcat[k][32..51] = Wr[k][:], rest 0.   [512,64]
// ---------------------------------------------------------------------------
__global__ void k_padW(const float* __restrict__ Wi, const float* __restrict__ Wr,
                       float* __restrict__ Wcat) {
    int idx = blockIdx.x * blockDim.x + threadIdx.x;   // [0, 512*64)
    int k = idx >> 6, c = idx & 63;
    float v = 0.f;
    if (c < NH)                    v = Wi[k * NH + c];
    else if (c >= 32 && c < 32+NH) v = Wr[k * NH + (c - 32)];
    Wcat[idx] = v;
}

// ---------------------------------------------------------------------------
// K2c (f=512): WMMA feature transform.  Y = X[32768,512] @ Wcat[512,64];
//   S[row,0..31] = dinv[row]*Y[row,0..31] (Wi part),  R[row,0..31] = Y[row,32..63].
// A-frag from LDS X tile (row stride 68 dwords -> banks 4L..4L+3, conflict-free
// b64 loads). B-frag from pair-interleaved LDS Wcat (pair stride 160 dwords,
// 160 mod 64 == 32 -> half-waves hit disjoint banks).
// ---------------------------------------------------------------------------
__global__ __launch_bounds__(256)
void k_xform_wmma(const float* __restrict__ X, const float* __restrict__ Wcat,
                  const float* __restrict__ dinv,
                  float* __restrict__ S, float* __restrict__ R) {
    constexpr int XS = 68;          // LDS X row stride (dwords)
    constexpr int PW = 160;         // LDS Wcat pair-row stride (dwords)
    __shared__ float ldsX[128 * XS];   //  8704 floats
    __shared__ float ldsW[32 * PW];    //  5120 floats  (total ~54 KB)

    const int tid  = threadIdx.x;
    const int w    = tid >> 5;
    const int lane = tid & 31;
    const int l16  = lane & 15;
    const int hs   = (lane < 16) ? 0 : 2;
    const int m0   = blockIdx.x * 128;          // block row base
    const int mrow = w * 16;                    // wave's local row tile

    v8f c[4] = {};

    for (int kc = 0; kc < 512; kc += 64) {
        // stage X[m0..m0+128) x [kc..kc+64) (coalesced)
        for (int e = tid; e < 128 * 64; e += 256) {
            int r = e >> 6, cc = e & 63;
            ldsX[r * XS + cc] = X[((size_t)m0 + r) * 512 + kc + cc];
        }
        // stage Wcat[kc..kc+64) x 64, pair-interleaved
        for (int e = tid; e < 64 * 64; e += 256) {
            int r = e >> 6, cc = e & 63;
            ldsW[(r >> 1) * PW + cc * 2 + (r & 1)] = Wcat[(size_t)(kc + r) * 64 + cc];
        }
        __syncthreads();

        #pragma unroll 2
        for (int k4 = 0; k4 < 64; k4 += 4) {
            v2f a = *(const v2f*)&ldsX[(mrow + l16) * XS + k4 + hs];
            const int wp = ((k4 + hs) >> 1) * PW;
            #pragma unroll
            for (int t = 0; t < 4; ++t) {
                v2f bb = *(const v2f*)&ldsW[wp + (t * 16 + l16) * 2];
                c[t] = __builtin_amdgcn_wmma_f32_16x16x4_f32(
                    false, a, false, bb, (short)0, c[t], false, false);
            }
        }
        __syncthreads();
    }

    // epilogue: vgpr r -> row mrow+r (lanes<16) / mrow+8+r (lanes>=16), col l16
    const int rbase = m0 + mrow + ((lane < 16) ? 0 : 8);
    #pragma unroll
    for (int r = 0; r < 8; ++r) {
        const int row = rbase + r;
        const float dv = dinv[row];
        S[(size_t)row * 32 +      l16] = dv * c[0][r];
        S[(size_t)row * 32 + 16 + l16] = dv * c[1][r];
        R[(size_t)row * 32 +      l16] = c[2][r];
        R[(size_t)row * 32 + 16 + l16] = c[3][r];
    }
}

// ---------------------------------------------------------------------------
// K3: WMMA propagate + epilogue.
//   P[b,j,h] = sum_i A[b,i,j] * S[b,i,h]          (A^T @ S per batch)
//   z        = dinv[b,j]*P + R[b,j,h] + bias[h]
//   out      = ACT==0 ? elu(gelu(z)) : relu(z)
// S staged pair-interleaved in LDS: one b64 ds load per B-frag, bank-disjoint
// half-waves (PW mod 64 == 32).
// ---------------------------------------------------------------------------
template <int NT, int ACT>
__global__ __launch_bounds__(256)
void k_prop(const float* __restrict__ A, const float* __restrict__ S,
            const float* __restrict__ R, const float* __restrict__ dinv,
            const float* __restrict__ bias, float* __restrict__ out,
            int Hused, int outStride) {
    constexpr int SC = 16 * NT;                 // global column stride of S / R
    constexpr int PW = (NT == 1) ? 32 : 96;     // LDS pair-row stride (dwords)
    __shared__ float lds[64 * PW];

    const int b    = blockIdx.x;
    const int jg   = blockIdx.y;
    const int tid  = threadIdx.x;
    const int w    = tid >> 5;
    const int lane = tid & 31;
    const int l16  = lane & 15;
    const int hs   = (lane < 16) ? 0 : 2;       // half-wave K offset
    const int j0   = (jg * 8 + w) * 16;

    v8f c[NT] = {};

    const float* Sb    = S + (size_t)b * NN * SC;
    const float* Abase = A + (size_t)b * NN * NN + (size_t)hs * NN + j0 + l16;

    for (int kb = 0; kb < 4; ++kb) {
        // stage S rows [kb*128, kb*128+128) pair-interleaved (coalesced reads)
        for (int e = tid; e < 128 * SC; e += 256) {
            int r  = e / SC;
            int cc = e % SC;
            lds[(r >> 1) * PW + cc * 2 + (r & 1)] =
                Sb[(size_t)(kb * 128 + r) * SC + cc];
        }
        __syncthreads();

        #pragma unroll 2
        for (int il = 0; il < 128; il += 4) {
            const int i0 = kb * 128 + il;
            v2f a;
            a.x = Abase[(size_t)i0 * NN];
            a.y = Abase[(size_t)(i0 + 1) * NN];
            const int wp = ((il + hs) >> 1) * PW;
            #pragma unroll
            for (int t = 0; t < NT; ++t) {
                v2f bb = *(const v2f*)&lds[wp + (t * 16 + l16) * 2];
                c[t] = __builtin_amdgcn_wmma_f32_16x16x4_f32(
                    false, a, false, bb, (short)0, c[t], false, false);
            }
        }
        __syncthreads();
    }

    // epilogue: D vgpr r -> row j0+r (lanes<16) / j0+8+r (lanes>=16), col = l16
    const int jbase = j0 + ((lane < 16) ? 0 : 8);
    #pragma unroll
    for (int r = 0; r < 8; ++r) {
        const int j = jbase + r;
        const float dvj = dinv[b * NN + j];
        #pragma unroll
        for (int t = 0; t < NT; ++t) {
            const int h = t * 16 + l16;
            if (h < Hused) {
                float z = dvj * c[t][r] + R[((size_t)b * NN + j) * SC + h] + bias[h];
                if (ACT == 0) {
                    float g = 0.5f * z * (1.0f + erff(z * 0.7071067811865475f));
                    z = (g > 0.f) ? g : (expf(g) - 1.0f);          // elu(gelu(z))
                } else {
                    z = fmaxf(z, 0.f);                             // relu
                }
                out[((size_t)b * NN + j) * outStride + h] = z;
            }
        }
    }
}

// ---------------------------------------------------------------------------
// K4: in-place row softmax over 7 classes
// ---------------------------------------------------------------------------
__global__ void k_softmax7(float* __restrict__ buf, int rows) {
    int r = blockIdx.x * blockDim.x + threadIdx.x;
    if (r >= rows) return;
    float* p = buf + (size_t)r * NC;
    float m = p[0];
    #pragma unroll
    for (int i = 1; i < NC; ++i) m = fmaxf(m, p[i]);
    float e[NC], s = 0.f;
    #pragma unroll
    for (int i = 0; i < NC; ++i) { e[i] = expf(p[i] - m); s += e[i]; }
    float inv = 1.0f / s;
    #pragma unroll
    for (int i = 0; i < NC; ++i) p[i] = e[i] * inv;
}

__device__ __forceinline__ void softmax7_inline(float* v) {
    float m = v[0];
    #pragma unroll
    for (int i = 1; i < NC; ++i) m = fmaxf(m, v[i]);
    float s = 0.f;
    #pragma unroll
    for (int i = 0; i < NC; ++i) { v[i] = expf(v[i] - m); s += v[i]; }
    float inv = 1.0f / s;
    #pragma unroll
    for (int i = 0; i < NC; ++i) v[i] *= inv;
}

// ---------------------------------------------------------------------------
// K5: dense head + mixing layers + final combine (one thread per node row)
// ---------------------------------------------------------------------------
__global__ void k_final(const float* __restrict__ xlt,
                        const float* __restrict__ ot,  const float* __restrict__ otw,
                        const float* __restrict__ otwe, const float* __restrict__ od,
                        const float* __restrict__ odur, const float* __restrict__ olt,
                        const float* __restrict__ Wd1, const float* __restrict__ bd1,
                        const float* __restrict__ Wd2, const float* __restrict__ bd2,
                        const float* __restrict__ Wll, const float* __restrict__ bll,
                        const float* __restrict__ Wout, const float* __restrict__ bout,
                        float* __restrict__ out) {
    int r = blockIdx.x * blockDim.x + threadIdx.x;
    if (r >= BG * NN) return;
    const float* x = xlt + (size_t)r * 48;

    // hidden = relu(x @ Wd1 + bd1), 40 units
    float hdn[40];
    #pragma unroll
    for (int j = 0; j < 40; ++j) hdn[j] = bd1[j];
    for (int k = 0; k < 48; ++k) {
        float xv = x[k];
        #pragma unroll
        for (int j = 0; j < 40; ++j) hdn[j] += xv * Wd1[k * 40 + j];
    }
    #pragma unroll
    for (int j = 0; j < 40; ++j) hdn[j] = fmaxf(hdn[j], 0.f);

    // out_ll = softmax(hidden @ Wd2 + bd2)
    float ll[NC];
    #pragma unroll
    for (int c = 0; c < NC; ++c) {
        float a = bd2[c];
        for (int j = 0; j < 40; ++j) a += hdn[j] * Wd2[j * NC + c];
        ll[c] = a;
    }
    softmax7_inline(ll);

    // out_dense = softmax((2*out_ll + 2*out_lt) @ Wll + bll)
    float v[NC], dl[NC];
    #pragma unroll
    for (int c = 0; c < NC; ++c) v[c] = 2.f * ll[c] + 2.f * olt[(size_t)r * NC + c];
    #pragma unroll
    for (int c = 0; c < NC; ++c) {
        float a = bll[c];
        #pragma unroll
        for (int j = 0; j < NC; ++j) a += v[j] * Wll[j * NC + c];
        dl[c] = a;
    }
    softmax7_inline(dl);

    // out_gnn = softmax((t+tw+twe+d+dur+out_dense) @ Wout + bout)
    float g[NC], gl[NC];
    #pragma unroll
    for (int c = 0; c < NC; ++c)
        g[c] = ot[(size_t)r * NC + c] + otw[(size_t)r * NC + c] + otwe[(size_t)r * NC + c]
             + od[(size_t)r * NC + c] + odur[(size_t)r * NC + c] + dl[c];
    #pragma unroll
    for (int c = 0; c < NC; ++c) {
        float a = bout[c];
        #pragma unroll
        for (int j = 0; j < NC; ++j) a += g[j] * Wout[j * NC + c];
        gl[c] = a;
    }
    softmax7_inline(gl);

    #pragma unroll
    for (int c = 0; c < NC; ++c) out[(size_t)r * NC + c] = dl[c] + gl[c];
}

// ---------------------------------------------------------------------------
// Launch
// ---------------------------------------------------------------------------
extern "C" void kernel_launch(void* const* d_in, const int* in_sizes, int n_in,
                              void* d_out, int out_size, void* d_ws, size_t ws_size,
                              hipStream_t stream) {
    auto F = [&](int i) { return (const float*)d_in[i]; };
    const int BN = BG * NN;

    float* w    = (float*)d_ws;
    float* dinv = w;  w += 4 * BN;
    float* S1   = w;  w += (size_t)BN * 32;
    float* R1   = w;  w += (size_t)BN * 32;
    float* Hb   = w;  w += (size_t)BN * NH;
    float* S2   = w;  w += (size_t)BN * 16;
    float* R2   = w;  w += (size_t)BN * 16;
    float* Wcat = w;  w += 512 * 64;
    float* outs[6];
    for (int i = 0; i < 6; ++i) { outs[i] = w; w += (size_t)BN * NC; }

    // degrees / dinv for the 4 adjacency matrices
    k_dinv<<<(4 * BN) / 256, 256, 0, stream>>>(F(0), F(1), F(2), F(3), dinv);

    // branch table: slot (output order), x input idx, weight-base idx,
    // adjacency idx m (== dinv segment), feature dim f.
    // t,d,dur first so A (64 MB) stays hot in the 192 MB L2 across 6 passes.
    struct Br { int slot, xi, wb, m, f; };
    const Br br[6] = {
        {0, 4, 10, 0, 48},   // t    (A)
        {3, 7, 28, 0, 512},  // d    (A)
        {4, 8, 34, 0, 512},  // dur  (A)
        {1, 5, 16, 1, 48},   // tw   (A_week)
        {2, 6, 22, 2, 48},   // twe  (A_weekend)
        {5, 9, 40, 3, 48},   // lt   (A_loc)
    };

    for (int q = 0; q < 6; ++q) {
        const Br& e = br[q];
        const float* A  = F(e.m);
        const float* dv = dinv + e.m * BN;
        // layer 1: f -> H=20 (padded to 32), act = elu(gelu(.))
        if (e.f == 512) {
            k_padW<<<(512 * 64) / 256, 256, 0, stream>>>(F(e.wb + 0), F(e.wb + 1), Wcat);
            k_xform_wmma<<<BN / 128, 256, 0, stream>>>(F(e.xi), Wcat, dv, S1, R1);
        } else {
            k_xform<<<(BN * 32) / 256, 256, 0, stream>>>(
                F(e.xi), F(e.wb + 0), F(e.wb + 1), dv, S1, R1, e.f, NH, 32);
        }
        k_prop<2, 0><<<dim3(BG, 4), 256, 0, stream>>>(
            A, S1, R1, dv, F(e.wb + 2), Hb, NH, NH);
        // layer 2: 20 -> C=7 (padded to 16), act = relu
        k_xform<<<(BN * 16) / 256, 256, 0, stream>>>(
            Hb, F(e.wb + 3), F(e.wb + 4), dv, S2, R2, NH, NC, 16);
        k_prop<1, 1><<<dim3(BG, 4), 256, 0, stream>>>(
            A, S2, R2, dv, F(e.wb + 5), outs[e.slot], NC, NC);
        k_softmax7<<<BN / 256, 256, 0, stream>>>(outs[e.slot], BN);
    }

    k_final<<<BN / 256, 256, 0, stream>>>(
        F(9), outs[0], outs[1], outs[2], outs[3], outs[4], outs[5],
        F(46), F(47), F(48), F(49), F(50), F(51), F(52), F(53), (float*)d_out);
}